// WindowAttention3D_15728170238469
// MI455X (gfx1250) — compile-verified
//
#include <hip/hip_runtime.h>
#include <hip/hip_bf16.h>
#include <math.h>

typedef __attribute__((ext_vector_type(2))) float v2f;
typedef __attribute__((ext_vector_type(8))) float v8f;

__device__ __forceinline__ v8f wmma_f32(v2f a, v2f b, v8f c) {
  // D = A(16x4) * B(4x16) + C, full fp32
  return __builtin_amdgcn_wmma_f32_16x16x4_f32(false, a, false, b, (short)0, c,
                                               false, false);
}

// ---------------------------------------------------------------------------
// Kernel 0: materialize relative-position bias  bias_pre[h][m][n] (H=4, N=98)
// ---------------------------------------------------------------------------
__global__ void bias_pre_kernel(const float* __restrict__ table,
                                float* __restrict__ bias_pre) {
  int tid = blockIdx.x * blockDim.x + threadIdx.x;
  if (tid >= 4 * 98 * 98) return;
  int h = tid / 9604;
  int rem = tid - h * 9604;
  int m = rem / 98;
  int n = rem - m * 98;
  int zm = m / 49, ym = (m % 49) / 7, xm = m % 7;
  int zn = n / 49, yn = (n % 49) / 7, xn = n % 7;
  int idx = (zm - zn + 1) * 169 + (ym - yn + 6) * 13 + (xm - xn + 6);
  bias_pre[tid] = table[idx * 4 + h];
}

// ---------------------------------------------------------------------------
// Fully fused window-attention: one workgroup per window (2048 blocks).
// 256 threads = 8 waves. All intermediates live in LDS (289.5 KB / 320 KB).
// Stage 1: qkv = x@w_qkv (+bias, q scaled)        -> sQKV[112][386]
// Stage 2: per (head, row-tile): S=QK^T +bias+mask, softmax, O=P@V -> sO
// Stage 3: out = sO@w_proj + b_proj               -> global
// ---------------------------------------------------------------------------
__global__ __launch_bounds__(256) void fused_window_attn(
    const float* __restrict__ x, const float* __restrict__ mask,
    const float* __restrict__ w_qkv, const float* __restrict__ bq,
    const float* __restrict__ w_proj, const float* __restrict__ bpj,
    const float* __restrict__ bias_pre, float* __restrict__ out) {
  // LDS (strides padded for bank-conflict-free WMMA-layout access)
  __shared__ float sQKV[112][386];      // rows 98..111 = clamped copies
  __shared__ float sO[112][130];        // pre-projection attention output
  __shared__ float sP[8][16][114];      // per-wave P staging (D->A relayout)

  const int b = blockIdx.x;             // window
  const int wd = b & 511;               // mask index (b % nW)
  const int wave = threadIdx.x >> 5;
  const int lane = threadIdx.x & 31;
  const int l16 = lane & 15;
  const int hi = lane >> 4;
  const float* xb = x + (size_t)b * 98 * 128;

  // ======== Stage 1: QKV projection into LDS ========
  // wave owns 3 N-tiles of 24; B regs reused across all 7 M-tiles.
  for (int i = 0; i < 3; ++i) {
    const int nt = wave * 3 + i;
    const int col = nt * 16 + l16;
    v8f acc[7] = {};
    for (int k0 = 0; k0 < 128; k0 += 4) {
      const int kb = k0 + 2 * hi;
      v2f bw;
      bw.x = w_qkv[(size_t)kb * 384 + col];
      bw.y = w_qkv[(size_t)(kb + 1) * 384 + col];
#pragma unroll
      for (int mt = 0; mt < 7; ++mt) {
        int n = mt * 16 + l16;
        int nc = n < 98 ? n : 97;                 // clamp padding rows
        v2f a = *(const v2f*)(xb + (size_t)nc * 128 + k0 + 2 * hi);
        acc[mt] = wmma_f32(a, bw, acc[mt]);
      }
    }
    const float bias = bq[col];
    const float scale = (col < 128) ? 0.17677669529663687f : 1.0f;
#pragma unroll
    for (int mt = 0; mt < 7; ++mt)
#pragma unroll
      for (int r = 0; r < 8; ++r) {
        int m = mt * 16 + r + 8 * hi;             // D-layout row
        sQKV[m][col] = (acc[mt][r] + bias) * scale;
      }
  }
  __syncthreads();

  // ======== Stage 2: attention, 28 units = 4 heads x 7 row-tiles ========
  for (int it = 0; it < 4; ++it) {
    const int u = wave + 8 * it;
    if (u < 28) {                                 // uniform per wave
      const int h = u / 7;
      const int mt = u - h * 7;

      // ---- S = Q K^T : M=16, N=112 (7 tiles), K=32 (8 steps) ----
      v8f acc[7] = {};
      for (int ks = 0; ks < 8; ++ks) {
        const int k0 = ks * 4;
        v2f a = *(const v2f*)(&sQKV[mt * 16 + l16][h * 32 + k0 + 2 * hi]);
#pragma unroll
        for (int nt = 0; nt < 7; ++nt) {
          v2f bk =
              *(const v2f*)(&sQKV[nt * 16 + l16][128 + h * 32 + k0 + 2 * hi]);
          acc[nt] = wmma_f32(a, bk, acc[nt]);
        }
      }

      // ---- + relative-position bias + window mask; -inf padding ----
      const float* bp = bias_pre + (size_t)h * 9604;
      const float* mp = mask + (size_t)wd * 9604;
#pragma unroll
      for (int nt = 0; nt < 7; ++nt) {
        const int n = nt * 16 + l16;
#pragma unroll
        for (int r = 0; r < 8; ++r) {
          const int m = mt * 16 + r + 8 * hi;
          if (n < 98 && m < 98)
            acc[nt][r] += bp[m * 98 + n] + mp[m * 98 + n];
          else
            acc[nt][r] = -3.0e38f;
        }
      }

      // ---- softmax per row (row lives in 16 lanes of a half-wave) ----
#pragma unroll
      for (int r = 0; r < 8; ++r) {
        float mx = acc[0][r];
#pragma unroll
        for (int nt = 1; nt < 7; ++nt) mx = fmaxf(mx, acc[nt][r]);
#pragma unroll
        for (int off = 1; off < 16; off <<= 1)
          mx = fmaxf(mx, __shfl_xor(mx, off, 32));
        float e[7];
        float sum = 0.0f;
#pragma unroll
        for (int nt = 0; nt < 7; ++nt) {
          e[nt] = __expf(acc[nt][r] - mx);
          sum += e[nt];
        }
#pragma unroll
        for (int off = 1; off < 16; off <<= 1) sum += __shfl_xor(sum, off, 32);
        const float inv = 1.0f / sum;
        const int m = mt * 16 + r + 8 * hi;
        const bool rowv = (m < 98);
#pragma unroll
        for (int nt = 0; nt < 7; ++nt) {
          const int n = nt * 16 + l16;
          sP[wave][r + 8 * hi][nt * 16 + l16] =
              (rowv && n < 98) ? e[nt] * inv : 0.0f;
        }
      }
      // same-wave LDS ops are ordered: no block barrier needed here

      // ---- O = P V : M=16, K=112 (28 steps), N=32 (2 tiles) ----
      v8f o0 = {}, o1 = {};
      for (int ks = 0; ks < 28; ++ks) {
        const int k0 = ks * 4;
        v2f a = *(const v2f*)(&sP[wave][l16][k0 + 2 * hi]);
        const int t0 = k0 + 2 * hi;               // P==0 for t>=98 (finite V)
        v2f b0, b1;
        b0.x = sQKV[t0][256 + h * 32 + l16];
        b0.y = sQKV[t0 + 1][256 + h * 32 + l16];
        o0 = wmma_f32(a, b0, o0);
        b1.x = sQKV[t0][256 + h * 32 + 16 + l16];
        b1.y = sQKV[t0 + 1][256 + h * 32 + 16 + l16];
        o1 = wmma_f32(a, b1, o1);
      }
#pragma unroll
      for (int r = 0; r < 8; ++r) {
        const int m = mt * 16 + r + 8 * hi;
        sO[m][h * 32 + l16] = o0[r];
        sO[m][h * 32 + 16 + l16] = o1[r];
      }
    }
  }
  __syncthreads();

  // ======== Stage 3: output projection sO[112][128] @ w_proj[128][128] ====
  {
    const int col = wave * 16 + l16;              // 8 N-tiles, one per wave
    v8f acc[7] = {};
    for (int k0 = 0; k0 < 128; k0 += 4) {
      const int kb = k0 + 2 * hi;
      v2f bw;
      bw.x = w_proj[(size_t)kb * 128 + col];
      bw.y = w_proj[(size_t)(kb + 1) * 128 + col];
#pragma unroll
      for (int mt = 0; mt < 7; ++mt) {
        v2f a = *(const v2f*)(&sO[mt * 16 + l16][k0 + 2 * hi]);
        acc[mt] = wmma_f32(a, bw, acc[mt]);
      }
    }
    const float bias = bpj[col];
#pragma unroll
    for (int mt = 0; mt < 7; ++mt)
#pragma unroll
      for (int r = 0; r < 8; ++r) {
        const int m = mt * 16 + r + 8 * hi;
        if (m < 98) out[((size_t)b * 98 + m) * 128 + col] = acc[mt][r] + bias;
      }
  }
}

// ---------------------------------------------------------------------------
extern "C" void kernel_launch(void* const* d_in, const int* in_sizes, int n_in,
                              void* d_out, int out_size, void* d_ws,
                              size_t ws_size, hipStream_t stream) {
  const float* x      = (const float*)d_in[0];
  const float* mask   = (const float*)d_in[1];
  const float* w_qkv  = (const float*)d_in[2];
  const float* b_qkv  = (const float*)d_in[3];
  const float* w_proj = (const float*)d_in[4];
  const float* b_proj = (const float*)d_in[5];
  const float* table  = (const float*)d_in[6];
  float* out = (float*)d_out;
  float* bias_pre = (float*)d_ws;      // [4,98,98] = 150 KB, L2-resident

  bias_pre_kernel<<<(4 * 98 * 98 + 255) / 256, 256, 0, stream>>>(table,
                                                                 bias_pre);
  fused_window_attn<<<2048, 256, 0, stream>>>(x, mask, w_qkv, b_qkv, w_proj,
                                              b_proj, bias_pre, out);
}